// GATRegressor_5085241279115
// MI455X (gfx1250) — compile-verified
//
#include <hip/hip_runtime.h>
#include <hip/hip_bf16.h>

#define N_NODES   100000
#define E_EDGES   1600000
#define IN_DIMV   128
#define HIDV      64
#define NEG_SLOPE 0.2f

typedef __attribute__((ext_vector_type(2))) float v2f;
typedef __attribute__((ext_vector_type(8))) float v8f;

// ---- order-preserving uint encoding of float (for deterministic atomic max) ----
__device__ __forceinline__ unsigned f2ord(float f) {
    unsigned u = __float_as_uint(f);
    return (u & 0x80000000u) ? ~u : (u | 0x80000000u);
}
__device__ __forceinline__ float ord2f(unsigned o) {
    return __uint_as_float((o & 0x80000000u) ? (o & 0x7FFFFFFFu) : ~o);
}

// ---------------------------------------------------------------------------
// Hout[N,64] = Hin[N,K] @ W[K,64] using V_WMMA_F32_16X16X4_F32 (fp32 tensor op).
// grid.x = N/16 row tiles; blockDim = 128 (4 waves -> 4 column tiles of 16).
// Wave32 layouts per CDNA5 ISA 7.12.2:
//   A 16x4 : lane m=lane%16, VGPR0/1 hold K = kb, kb+1 with kb = 2*(lane/16)
//   B 4x16 : lane n=lane%16, VGPR0/1 hold rows K = kb, kb+1
//   C/D    : VGPR j -> row (j + 8*(lane/16)), col = lane%16
// EXEC must be all ones: no divergence before/around the WMMA (exact grid).
// ---------------------------------------------------------------------------
template <int K>
__global__ void __launch_bounds__(128)
gemm_wmma_f32(const float* __restrict__ Hin, const float* __restrict__ W,
              float* __restrict__ Hout)
{
    const int lane = threadIdx.x & 31;
    const int wv   = threadIdx.x >> 5;           // column tile 0..3
    const int row0 = blockIdx.x << 4;
    const int col0 = wv << 4;
    const int m    = lane & 15;                  // row within A tile / col within B tile
    const int kb   = (lane >> 4) << 1;           // 0 or 2

    v8f acc = {};
    const float* arow = Hin + (size_t)(row0 + m) * K;
    #pragma unroll 4
    for (int k0 = 0; k0 < K; k0 += 4) {
        v2f a, b;
        a.x = arow[k0 + kb];
        a.y = arow[k0 + kb + 1];
        b.x = W[(size_t)(k0 + kb)     * HIDV + col0 + m];
        b.y = W[(size_t)(k0 + kb + 1) * HIDV + col0 + m];
        acc = __builtin_amdgcn_wmma_f32_16x16x4_f32(
                  /*neg_a=*/false, a, /*neg_b=*/false, b,
                  /*c_mod=*/(short)0, acc, /*reuse_a=*/false, /*reuse_b=*/false);
    }

    const int rbase = row0 + ((lane >> 4) << 3);
    #pragma unroll
    for (int j = 0; j < 8; ++j)
        Hout[(size_t)(rbase + j) * HIDV + col0 + m] = acc[j];
}

// ---- per-node attention logits: as[n]=<h[n],a_src>, ad[n]=<h[n],a_dst> -----
__global__ void __launch_bounds__(256)
node_alpha(const float* __restrict__ H, const float* __restrict__ avs,
           const float* __restrict__ avd, float* __restrict__ as_,
           float* __restrict__ ad_, int n)
{
    const int lane = threadIdx.x & 31;
    const int node = (int)((blockIdx.x * (size_t)blockDim.x + threadIdx.x) >> 5);
    if (node >= n) return;
    const float* h = H + (size_t)node * HIDV;
    float s = h[lane] * avs[lane] + h[lane + 32] * avs[lane + 32];
    float d = h[lane] * avd[lane] + h[lane + 32] * avd[lane + 32];
    #pragma unroll
    for (int off = 16; off; off >>= 1) {
        s += __shfl_xor(s, off, 32);
        d += __shfl_xor(d, off, 32);
    }
    if (lane == 0) { as_[node] = s; ad_[node] = d; }
}

// ---- init per-layer node state: agg=0, m=-inf (ordered), denom=0 -----------
__global__ void __launch_bounds__(256)
init_node(unsigned* __restrict__ mOrd, float* __restrict__ den, float* __restrict__ agg)
{
    const size_t i = blockIdx.x * (size_t)blockDim.x + threadIdx.x;
    if (i >= (size_t)N_NODES * HIDV) return;
    agg[i] = 0.0f;
    if (i < N_NODES) { mOrd[i] = f2ord(-__builtin_inff()); den[i] = 0.0f; }
}

// ---- pass 1: leaky-relu logit per edge + segment max over dst --------------
__global__ void __launch_bounds__(256)
edge_logits(const int* __restrict__ se, const int* __restrict__ de,
            const float* __restrict__ as_, const float* __restrict__ ad_,
            float* __restrict__ eT, unsigned* __restrict__ mOrd)
{
    const int ET = E_EDGES + N_NODES;
    const int i = (int)(blockIdx.x * (size_t)blockDim.x + threadIdx.x);
    if (i >= ET) return;
    int s, d;
    if (i < E_EDGES) { s = se[i]; d = de[i]; } else { s = d = i - E_EDGES; }
    float ev = as_[s] + ad_[d];
    ev = ev > 0.0f ? ev : NEG_SLOPE * ev;
    eT[i] = ev;
    atomicMax(&mOrd[d], f2ord(ev));
}

// ---- pass 2: exp(e - max) + segment sum over dst ---------------------------
__global__ void __launch_bounds__(256)
edge_exp(const int* __restrict__ de, float* __restrict__ eT,
         const unsigned* __restrict__ mOrd, float* __restrict__ den)
{
    const int ET = E_EDGES + N_NODES;
    const int i = (int)(blockIdx.x * (size_t)blockDim.x + threadIdx.x);
    if (i >= ET) return;
    const int d = (i < E_EDGES) ? de[i] : (i - E_EDGES);
    const float ex = expf(eT[i] - ord2f(mOrd[d]));
    eT[i] = ex;
    atomicAdd(&den[d], ex);
}

// ---- pass 3: wave-per-edge gather h[src], scatter-add alpha*h into agg[dst]
__global__ void __launch_bounds__(256)
edge_scatter(const int* __restrict__ se, const int* __restrict__ de,
             const float* __restrict__ H, const float* __restrict__ eT,
             const float* __restrict__ den, float* __restrict__ agg)
{
    const int ET = E_EDGES + N_NODES;
    const int lane = threadIdx.x & 31;
    const int e = (int)(blockIdx.x * (size_t)(blockDim.x >> 5) + (threadIdx.x >> 5));
    if (e >= ET) return;
    int s, d;
    if (e < E_EDGES) { s = se[e]; d = de[e]; } else { s = d = e - E_EDGES; }
    const float alpha = eT[e] / den[d];
    const float* hs = H + (size_t)s * HIDV;
    float* ag = agg + (size_t)d * HIDV;
    atomicAdd(&ag[lane],      hs[lane]      * alpha);
    atomicAdd(&ag[lane + 32], hs[lane + 32] * alpha);
}

// ---- bias + ELU ------------------------------------------------------------
__global__ void __launch_bounds__(256)
finish_elu(const float* __restrict__ agg, const float* __restrict__ bias,
           float* __restrict__ out)
{
    const size_t i = blockIdx.x * (size_t)blockDim.x + threadIdx.x;
    if (i >= (size_t)N_NODES * HIDV) return;
    const float v = agg[i] + bias[i & (HIDV - 1)];
    out[i] = v > 0.0f ? v : expm1f(v);
}

// ---- final projection: y[n] = <h[n], W_out> + b_out ------------------------
__global__ void __launch_bounds__(256)
out_proj(const float* __restrict__ H, const float* __restrict__ Wo,
         const float* __restrict__ bo, float* __restrict__ y, int n)
{
    const int lane = threadIdx.x & 31;
    const int node = (int)((blockIdx.x * (size_t)blockDim.x + threadIdx.x) >> 5);
    if (node >= n) return;
    const float* h = H + (size_t)node * HIDV;
    float s = h[lane] * Wo[lane] + h[lane + 32] * Wo[lane + 32];
    #pragma unroll
    for (int off = 16; off; off >>= 1) s += __shfl_xor(s, off, 32);
    if (lane == 0) y[node] = s + bo[0];
}

extern "C" void kernel_launch(void* const* d_in, const int* in_sizes, int n_in,
                              void* d_out, int out_size, void* d_ws, size_t ws_size,
                              hipStream_t stream)
{
    const float* x      = (const float*)d_in[0];
    const int*   ei     = (const int*)d_in[1];     // [2,E] int32
    const float* W1     = (const float*)d_in[2];
    const float* a_src1 = (const float*)d_in[3];
    const float* a_dst1 = (const float*)d_in[4];
    const float* b1     = (const float*)d_in[5];
    const float* W2     = (const float*)d_in[6];
    const float* a_src2 = (const float*)d_in[7];
    const float* a_dst2 = (const float*)d_in[8];
    const float* b2     = (const float*)d_in[9];
    const float* W_out  = (const float*)d_in[10];
    const float* b_out  = (const float*)d_in[11];
    float* y = (float*)d_out;

    const int* se = ei;             // edge_index[0]
    const int* de = ei + E_EDGES;   // edge_index[1]

    // workspace carve-up (~85 MB total; fits MI455X 192 MB L2)
    float*    h1   = (float*)d_ws;                               // N*64
    float*    h2   = h1 + (size_t)N_NODES * HIDV;                // N*64
    float*    agg  = h2 + (size_t)N_NODES * HIDV;                // N*64
    float*    as_  = agg + (size_t)N_NODES * HIDV;               // N
    float*    ad_  = as_ + N_NODES;                              // N
    float*    den  = ad_ + N_NODES;                              // N
    unsigned* mOrd = (unsigned*)(den + N_NODES);                 // N
    float*    eT   = (float*)(mOrd + N_NODES);                   // E+N

    const int ET       = E_EDGES + N_NODES;
    const int gRowT    = N_NODES / 16;               // 6250 (exact)
    const int gNodeVec = (N_NODES * HIDV + 255) / 256;
    const int gNodeWav = (N_NODES + 7) / 8;
    const int gEdgeThr = (ET + 255) / 256;
    const int gEdgeWav = (ET + 7) / 8;

    // ---------------- layer 1 ----------------
    gemm_wmma_f32<IN_DIMV><<<gRowT, 128, 0, stream>>>(x, W1, h1);
    node_alpha<<<gNodeWav, 256, 0, stream>>>(h1, a_src1, a_dst1, as_, ad_, N_NODES);
    init_node<<<gNodeVec, 256, 0, stream>>>(mOrd, den, agg);
    edge_logits<<<gEdgeThr, 256, 0, stream>>>(se, de, as_, ad_, eT, mOrd);
    edge_exp<<<gEdgeThr, 256, 0, stream>>>(de, eT, mOrd, den);
    edge_scatter<<<gEdgeWav, 256, 0, stream>>>(se, de, h1, eT, den, agg);
    finish_elu<<<gNodeVec, 256, 0, stream>>>(agg, b1, h1);   // h1 <- elu(agg+b1)

    // ---------------- layer 2 ----------------
    gemm_wmma_f32<HIDV><<<gRowT, 128, 0, stream>>>(h1, W2, h2);
    node_alpha<<<gNodeWav, 256, 0, stream>>>(h2, a_src2, a_dst2, as_, ad_, N_NODES);
    init_node<<<gNodeVec, 256, 0, stream>>>(mOrd, den, agg);
    edge_logits<<<gEdgeThr, 256, 0, stream>>>(se, de, as_, ad_, eT, mOrd);
    edge_exp<<<gEdgeThr, 256, 0, stream>>>(de, eT, mOrd, den);
    edge_scatter<<<gEdgeWav, 256, 0, stream>>>(se, de, h2, eT, den, agg);
    finish_elu<<<gNodeVec, 256, 0, stream>>>(agg, b2, h2);   // h2 <- elu(agg+b2)

    // ---------------- output head ----------------
    out_proj<<<gNodeWav, 256, 0, stream>>>(h2, W_out, b_out, y, N_NODES);
}